// SCLSTMCellReleased_18906446037641
// MI455X (gfx1250) — compile-verified
//
#include <hip/hip_runtime.h>
#include <math.h>

typedef float v2f __attribute__((ext_vector_type(2)));
typedef float v8f __attribute__((ext_vector_type(8)));

#define ACT 128
#define SV  1024
#define IN_ 1024
#define H_  2048
#define G4  (4 * H_)        // 8192 gate rows
#define KU  (IN_ + H_ + SV) // 4096 combined input length
#define KD  (ACT + SV)      // 1152 dc input length

__device__ __forceinline__ float sigm(float x) { return 1.0f / (1.0f + expf(-x)); }

// Accumulate 16 output rows (row0..row0+15) of y += W[rows, kbeg:kend] * u[ubase+kbeg:ubase+kend]
// via V_WMMA_F32_16X16X4_F32. A: lane m holds W[row0+m, k..k+1], lane m+16 holds W[row0+m, k+2..k+3]
// (32-bit 16x4 A layout) -> one global_load_b64 per lane per quad. B: activation quad broadcast
// across N -> one ds_load_b64 at (ubase+k+sel). D[:,n] identical for all n; rows 0-7 in acc[0..7]
// of lanes 0-15, rows 8-15 in acc[0..7] of lanes 16-31. kbeg/kend must be wave-uniform SGPRs.
__device__ __forceinline__ v8f wmma_gemv_seg(const float* __restrict__ W, int ld, int row0,
                                             const float* su, int ubase,
                                             int kbeg, int kend, int m, int sel, v8f acc) {
  const float* wrow = W + (size_t)(row0 + m) * (size_t)ld + (size_t)sel;
  const float* ub   = su + ubase + sel;
#pragma unroll 8
  for (int k = kbeg; k < kend; k += 4) {
    v2f a = *(const v2f*)(wrow + k);  // global_load_b64, streaming per row
    v2f b = *(const v2f*)(ub + k);    // ds_load_b64 from staged activations
    acc = __builtin_amdgcn_wmma_f32_16x16x4_f32(false, a, false, b, (short)0, acc,
                                                false, false);
  }
  return acc;
}

// ---------------- Kernel 1: gates[8192] and read-gate preact[1024] -------------------
// block = 256 threads = 8 waves = 2 row-groups x 4 K-split waves. grid = 288 (576 groups).
__global__ void k_gates(const float* __restrict__ x, const float* __restrict__ hx,
                        const float* __restrict__ dx,
                        const float* __restrict__ Wih, const float* __restrict__ Whh,
                        const float* __restrict__ Wsvh,
                        const float* __restrict__ Wwr, const float* __restrict__ Whr,
                        const float* __restrict__ Wsvr,
                        float* __restrict__ gates, float* __restrict__ rpre) {
  __shared__ float su[KU];
  __shared__ float red[2][4][16];
  const int tid = threadIdx.x;

  for (int i = tid; i < IN_; i += 256) su[i] = x[i];
  for (int i = tid; i < H_; i += 256) su[IN_ + i] = hx[i];
  for (int i = tid; i < SV; i += 256) su[IN_ + H_ + i] = dx[ACT + i];
  __syncthreads();

  const int lane = tid & 31;
  // wave-uniform scalars -> force into SGPRs so K-loops are scalar-controlled
  const int wave = __builtin_amdgcn_readfirstlane(tid >> 5);
  const int lgrp = wave >> 2;              // local row-group (0..1)
  const int s    = wave & 3;               // K-split id (0..3)
  const int grp  = blockIdx.x * 2 + lgrp;  // 0..575
  const int m    = lane & 15;
  const int sel  = (lane >> 4) << 1;       // 0 or 2

  int row0;
  const float *W0, *W1, *W2;
  if (grp < G4 / 16) { row0 = grp * 16;              W0 = Wih; W1 = Whh; W2 = Wsvh; }
  else               { row0 = (grp - G4 / 16) * 16;  W0 = Wwr; W1 = Whr; W2 = Wsvr; }

  v8f acc = {0.f, 0.f, 0.f, 0.f, 0.f, 0.f, 0.f, 0.f};
  acc = wmma_gemv_seg(W0, IN_, row0, su, 0,        s * (IN_ / 4), (s + 1) * (IN_ / 4), m, sel, acc);
  acc = wmma_gemv_seg(W1, H_,  row0, su, IN_,      s * (H_  / 4), (s + 1) * (H_  / 4), m, sel, acc);
  acc = wmma_gemv_seg(W2, SV,  row0, su, IN_ + H_, s * (SV  / 4), (s + 1) * (SV  / 4), m, sel, acc);

  if ((lane & 15) == 0) {
    const int half = lane >> 4;
#pragma unroll
    for (int i = 0; i < 8; ++i) red[lgrp][s][half * 8 + i] = acc[i];
  }
  __syncthreads();

  if (tid < 32) {
    const int lg = tid >> 4, r = tid & 15;
    const int g2 = blockIdx.x * 2 + lg;
    const float v = red[lg][0][r] + red[lg][1][r] + red[lg][2][r] + red[lg][3][r];
    if (g2 < G4 / 16) gates[g2 * 16 + r] = v;
    else              rpre[(g2 - G4 / 16) * 16 + r] = v;
  }
}

// ---------------- Kernel 2: dy = [a0, sigmoid(rpre)*svx] (also dc input vector) ------
__global__ void k_dy(const float* __restrict__ dx, const float* __restrict__ rpre,
                     float* __restrict__ dy_out, float* __restrict__ dvec) {
  const int i = blockIdx.x * blockDim.x + threadIdx.x;
  if (i < KD) {
    float v = dx[i];
    if (i >= ACT) v = v * sigm(rpre[i - ACT]);
    dy_out[i] = v;
    dvec[i]   = v;
  }
}

// ---------------- Kernel 3: dc GEMV + fused LSTM elementwise -------------------------
// block = 256 = 2 groups x 4 K-split waves; grid = 64 (128 groups of 16 rows = 2048).
__global__ void k_cell(const float* __restrict__ dvec, const float* __restrict__ Wdc,
                       const float* __restrict__ gates, const float* __restrict__ cx,
                       float* __restrict__ out_hy, float* __restrict__ out_cy) {
  __shared__ float sd[KD];
  __shared__ float red[2][4][16];
  const int tid = threadIdx.x;
  for (int i = tid; i < KD; i += 256) sd[i] = dvec[i];
  __syncthreads();

  const int lane = tid & 31;
  const int wave = __builtin_amdgcn_readfirstlane(tid >> 5);
  const int lgrp = wave >> 2, s = wave & 3;
  const int grp  = blockIdx.x * 2 + lgrp;  // 0..127
  const int row0 = grp * 16;
  const int m    = lane & 15;
  const int sel  = (lane >> 4) << 1;

  v8f acc = {0.f, 0.f, 0.f, 0.f, 0.f, 0.f, 0.f, 0.f};
  acc = wmma_gemv_seg(Wdc, KD, row0, sd, 0, s * (KD / 4), (s + 1) * (KD / 4), m, sel, acc);

  if ((lane & 15) == 0) {
    const int half = lane >> 4;
#pragma unroll
    for (int i = 0; i < 8; ++i) red[lgrp][s][half * 8 + i] = acc[i];
  }
  __syncthreads();

  if (tid < 32) {
    const int lg = tid >> 4, r = tid & 15;
    const int j = (blockIdx.x * 2 + lg) * 16 + r;
    const float dc = tanhf(red[lg][0][r] + red[lg][1][r] + red[lg][2][r] + red[lg][3][r]);
    const float ig = sigm(gates[j]);
    const float fg = sigm(gates[H_ + j]);
    const float cg = tanhf(gates[2 * H_ + j]);
    const float og = sigm(gates[3 * H_ + j]);
    const float cy = fg * cx[j] + ig * cg + dc;
    out_hy[j] = og * tanhf(cy);
    out_cy[j] = cy;
  }
}

extern "C" void kernel_launch(void* const* d_in, const int* in_sizes, int n_in,
                              void* d_out, int out_size, void* d_ws, size_t ws_size,
                              hipStream_t stream) {
  const float* x    = (const float*)d_in[0];   // [1024]
  const float* hx   = (const float*)d_in[1];   // [2048]
  const float* cx   = (const float*)d_in[2];   // [2048]
  const float* dx   = (const float*)d_in[3];   // [1152]
  const float* Wih  = (const float*)d_in[4];   // [8192,1024]
  const float* Whh  = (const float*)d_in[5];   // [8192,2048]
  const float* Wsvh = (const float*)d_in[6];   // [8192,1024]
  const float* Wwr  = (const float*)d_in[7];   // [1024,1024]
  const float* Whr  = (const float*)d_in[8];   // [1024,2048]
  const float* Wsvr = (const float*)d_in[9];   // [1024,1024]
  const float* Wdc  = (const float*)d_in[10];  // [2048,1152]

  float* out = (float*)d_out;          // hy[0:2048] | cy[2048:4096] | dy[4096:5248]
  float* ws  = (float*)d_ws;
  float* gates = ws;                   // [8192]
  float* rpre  = ws + G4;              // [1024]
  float* dvec  = ws + G4 + SV;         // [1152]

  k_gates<<<dim3((G4 / 16 + SV / 16) / 2), dim3(256), 0, stream>>>(
      x, hx, dx, Wih, Whh, Wsvh, Wwr, Whr, Wsvr, gates, rpre);

  k_dy<<<dim3((KD + 255) / 256), dim3(256), 0, stream>>>(dx, rpre, out + 2 * H_, dvec);

  k_cell<<<dim3((H_ / 16) / 2), dim3(256), 0, stream>>>(
      dvec, Wdc, gates, cx, out, out + H_);
}